// Attn_67980742361180
// MI455X (gfx1250) — compile-verified
//
#include <hip/hip_runtime.h>
#include <math.h>

typedef __attribute__((ext_vector_type(2))) float v2f;
typedef __attribute__((ext_vector_type(8))) float v8f;

#define EDIM 512
#define LDIM 1024

// ---------------------------------------------------------------------------
// Phase 1: u[e] = sum_f W[f,e] * v[f]   (note einsum "ble,fe->blf" => W^T v)
//          c    = bias . v
// Tiny (1 MB read) -> single workgroup, plain VALU.
// ws[0..511] = u, ws[512] = c
// ---------------------------------------------------------------------------
__global__ __launch_bounds__(512)
void attn_prep(const float* __restrict__ W, const float* __restrict__ bias,
               const float* __restrict__ wvec, float* __restrict__ ws) {
    __shared__ float v_lds[EDIM];
    __shared__ float red[512];
    const int t = threadIdx.x;
    v_lds[t] = wvec[t];
    __syncthreads();
    float acc = 0.0f;
#pragma unroll 8
    for (int f = 0; f < EDIM; ++f)
        acc = fmaf(W[(size_t)f * EDIM + t], v_lds[f], acc);
    ws[t] = acc;
    red[t] = bias[t] * v_lds[t];
    __syncthreads();
    for (int s = 256; s > 0; s >>= 1) {
        if (t < s) red[t] += red[t + s];
        __syncthreads();
    }
    if (t == 0) ws[EDIM] = red[0];
}

// ---------------------------------------------------------------------------
// Phase 2: one block per batch row b.
//   scores[l] = Q[b,l,:] . u + c   via V_WMMA_F32_16X16X4_F32
//   then length-masked softmax over L in-block.
// A operand: u chunk in matrix row 0 only (lanes 0 / 16), rest zero.
// B operand: 16 tokens as columns -> D[0][n] = score of token n,
//            which lands in acc[0] of lanes 0..15 (C/D layout: VGPR0,
//            lanes 0-15 = M=0, N=lane).
// ---------------------------------------------------------------------------
__global__ __launch_bounds__(256)
void attn_main(const float* __restrict__ q, const int* __restrict__ lens,
               const float* __restrict__ ws, float* __restrict__ out) {
    __shared__ float u_lds[EDIM];
    __shared__ float scores[LDIM];
    __shared__ float red[256];

    const int tid  = threadIdx.x;
    const int lane = tid & 31;
    const int wave = tid >> 5;
    const int b    = blockIdx.x;

    // stage u (2 KB) into LDS once per block
    for (int i = tid; i < EDIM; i += 256) u_lds[i] = ws[i];
    __syncthreads();

    // 32-bit A/B operand layout for 16x16x4:
    //   lanes 0-15  -> K = {k, k+1}   (VGPR0, VGPR1)
    //   lanes 16-31 -> K = {k+2, k+3}
    const int  koff    = (lane >> 4) * 2;        // 0 or 2
    const bool contrib = ((lane & 15) == 0);     // lanes 0,16 carry A row 0
    const int  tok_in_tile = lane & 15;
    const v2f  zz = {0.0f, 0.0f};

    // 64 tiles of 16 tokens; 8 waves -> 8 tiles per wave
    for (int t = wave; t < LDIM / 16; t += 8) {
        const int tok = t * 16 + tok_in_tile;
        const float* qrow = q + ((size_t)b * LDIM + tok) * EDIM + koff;
        v8f acc = {0.0f, 0.0f, 0.0f, 0.0f, 0.0f, 0.0f, 0.0f, 0.0f};
#pragma unroll 8
        for (int k = 0; k < EDIM; k += 4) {
            v2f bq = *(const v2f*)(qrow + k);                 // global_load_b64
            v2f uu = *(const v2f*)(&u_lds[k + koff]);         // ds_load_b64
            v2f av = contrib ? uu : zz;                       // zero rows 1..15
            acc = __builtin_amdgcn_wmma_f32_16x16x4_f32(
                    /*neg_a=*/false, av, /*neg_b=*/false, bq,
                    /*c_mod=*/(short)0, acc,
                    /*reuse_a=*/false, /*reuse_b=*/false);
        }
        if (lane < 16) scores[t * 16 + lane] = acc[0];        // D[0][lane]
    }
    __syncthreads();

    // ---------------- masked softmax over L ----------------
    const int   qlen = lens[b];
    const float c    = ws[EDIM];

    float lmax = -INFINITY;
    for (int l = tid; l < LDIM; l += 256) {
        float s = (l < qlen) ? (scores[l] + c) : -INFINITY;
        scores[l] = s;
        lmax = fmaxf(lmax, s);
    }
    red[tid] = lmax;
    __syncthreads();
    for (int s = 128; s > 0; s >>= 1) {
        if (tid < s) red[tid] = fmaxf(red[tid], red[tid + s]);
        __syncthreads();
    }
    const float gmax = red[0];
    __syncthreads();

    float lsum = 0.0f;
    for (int l = tid; l < LDIM; l += 256) {
        float e = (l < qlen) ? expf(scores[l] - gmax) : 0.0f;
        scores[l] = e;
        lsum += e;
    }
    red[tid] = lsum;
    __syncthreads();
    for (int s = 128; s > 0; s >>= 1) {
        if (tid < s) red[tid] += red[tid + s];
        __syncthreads();
    }
    const float inv = 1.0f / red[0];

    for (int l = tid; l < LDIM; l += 256)
        out[(size_t)b * LDIM + l] = scores[l] * inv;
}

// ---------------------------------------------------------------------------
extern "C" void kernel_launch(void* const* d_in, const int* in_sizes, int n_in,
                              void* d_out, int out_size, void* d_ws, size_t ws_size,
                              hipStream_t stream) {
    const float* questions = (const float*)d_in[0];
    const int*   lens      = (const int*)d_in[1];
    const float* W         = (const float*)d_in[2];
    const float* bias      = (const float*)d_in[3];
    const float* wvec      = (const float*)d_in[4];
    float*       out       = (float*)d_out;
    float*       ws        = (float*)d_ws;   // needs (EDIM+1)*4 = 2052 bytes

    const int B2 = in_sizes[1];              // one block per batch row

    attn_prep<<<1, 512, 0, stream>>>(W, bias, wvec, ws);
    attn_main<<<B2, 256, 0, stream>>>(questions, lens, ws, out);
}